// PerTokenDistillationLoss_5231270166835
// MI455X (gfx1250) — compile-verified
//
#include <hip/hip_runtime.h>
#include <hip/hip_bf16.h>
#include <math.h>
#include <stdint.h>

// Problem geometry (fixed by the reference).
#define BATCH   2
#define SEQ     2048
#define VOCAB   32000
#define NPOS    (SEQ - 1)          // 2047 positions per batch row
#define NROWS   (BATCH * NPOS)     // 4094 token rows
#define TPB     320                // 10 waves of 32; 32000 = 320*4*25 exactly
#define ITERS   25                 // float4 iterations per tensor per row
#define CHUNKB  (TPB * 16)         // 5120 B per tensor chunk
#define STAGEB  (2 * CHUNKB)       // 10240 B per stage (student + teacher)
#define NSTAGE  4                  // async pipeline depth (4 * 10240 = 40960 B LDS)
#define IGNORE_INDEX (-100)

// ---- CDNA5 async global->LDS DMA (ASYNCcnt path), non-temporal stream ----
__device__ __forceinline__ void async_ld_b128(uint32_t lds_off, uint64_t gaddr) {
    // VDST = per-lane LDS byte address, VADDR = 64-bit global address (GV mode)
    asm volatile("global_load_async_to_lds_b128 %0, %1, off th:TH_LOAD_NT"
                 :: "v"(lds_off), "v"(gaddr) : "memory");
}

__global__ __launch_bounds__(TPB)
void distill_row_kernel(const int* __restrict__ gt,
                        const float* __restrict__ slog,
                        const float* __restrict__ tlog,
                        float* __restrict__ ws) {
    __shared__ __align__(16) unsigned char stage[NSTAGE * STAGEB]; // 40960 B
    __shared__ float red[TPB / 32][6];

    const int r = blockIdx.x;          // token row 0..4093
    const int b = r / NPOS;
    const int s = r % NPOS;
    const size_t row = ((size_t)b * SEQ + (size_t)s) * (size_t)VOCAB;
    const float* sp = slog + row;
    const float* tp = tlog + row;

    const uint32_t tid   = threadIdx.x;
    const uint32_t lbase = (uint32_t)(uintptr_t)(&stage[0]); // generic->LDS offset (low 32 bits)
    const uint32_t slot  = tid * 16u;

    // ---- prologue: fill NSTAGE-1 stages ------------------------------------
    #pragma unroll
    for (int k = 0; k < NSTAGE - 1; ++k) {
        const uint32_t lo  = lbase + (uint32_t)k * STAGEB + slot;
        const size_t   off = (size_t)k * (TPB * 4) + tid * 4;
        async_ld_b128(lo,          (uint64_t)(uintptr_t)(sp + off));
        async_ld_b128(lo + CHUNKB, (uint64_t)(uintptr_t)(tp + off));
    }

    // per-thread online stats
    float m_s = -INFINITY, Z_s = 0.0f;
    float m_t = -INFINITY, Z_t = 0.0f, A = 0.0f, Bm = 0.0f;

    for (int i = 0; i < ITERS; ++i) {
        const uint32_t p = (uint32_t)(i & (NSTAGE - 1));
        if (i + NSTAGE - 1 < ITERS) {
            // buffer (i+NSTAGE-1)&3 was last read at iteration i-1: ensure those
            // ds reads fully retired before the DMA may overwrite it.
            asm volatile("s_wait_dscnt 0x0" ::: "memory");
            const int      nx = i + NSTAGE - 1;
            const uint32_t pn = (uint32_t)(nx & (NSTAGE - 1));
            const uint32_t lo = lbase + pn * STAGEB + slot;
            const size_t  off = (size_t)nx * (TPB * 4) + tid * 4;
            async_ld_b128(lo,          (uint64_t)(uintptr_t)(sp + off));
            async_ld_b128(lo + CHUNKB, (uint64_t)(uintptr_t)(tp + off));
        }
        // Async loads complete in order: outstanding <= 2*(stages beyond i)
        // guarantees stage i's two transfers have landed in LDS.
        const int rem = ITERS - 1 - i;  // unread stages after this one
        if (rem >= NSTAGE - 1)      asm volatile("s_wait_asynccnt 0x6" ::: "memory");
        else if (rem == 2)          asm volatile("s_wait_asynccnt 0x4" ::: "memory");
        else if (rem == 1)          asm volatile("s_wait_asynccnt 0x2" ::: "memory");
        else                        asm volatile("s_wait_asynccnt 0x0" ::: "memory");

        const float4 sv = *(const float4*)(&stage[p * STAGEB + slot]);
        const float4 tv = *(const float4*)(&stage[p * STAGEB + CHUNKB + slot]);

        // student logsumexp (online, one rescale per 4 elems)
        {
            const float c4 = fmaxf(fmaxf(sv.x, sv.y), fmaxf(sv.z, sv.w));
            const float nm = fmaxf(m_s, c4);
            Z_s = Z_s * __expf(m_s - nm)
                + (__expf(sv.x - nm) + __expf(sv.y - nm)
                 + __expf(sv.z - nm) + __expf(sv.w - nm));
            m_s = nm;
        }
        // teacher logsumexp + KL partials
        {
            const float c4 = fmaxf(fmaxf(tv.x, tv.y), fmaxf(tv.z, tv.w));
            const float nm = fmaxf(m_t, c4);
            const float rs = __expf(m_t - nm);
            const float e0 = __expf(tv.x - nm), e1 = __expf(tv.y - nm);
            const float e2 = __expf(tv.z - nm), e3 = __expf(tv.w - nm);
            // inf-mask on student logits (select, so no 0*inf NaN)
            const float a0 = isinf(sv.x) ? 0.0f : e0 * (tv.x - sv.x);
            const float a1 = isinf(sv.y) ? 0.0f : e1 * (tv.y - sv.y);
            const float a2 = isinf(sv.z) ? 0.0f : e2 * (tv.z - sv.z);
            const float a3 = isinf(sv.w) ? 0.0f : e3 * (tv.w - sv.w);
            const float b0 = isinf(sv.x) ? 0.0f : e0;
            const float b1 = isinf(sv.y) ? 0.0f : e1;
            const float b2 = isinf(sv.z) ? 0.0f : e2;
            const float b3 = isinf(sv.w) ? 0.0f : e3;
            Z_t = Z_t * rs + (e0 + e1 + e2 + e3);
            A   = A   * rs + (a0 + a1 + a2 + a3);
            Bm  = Bm  * rs + (b0 + b1 + b2 + b3);
            m_t = nm;
        }
    }

    // ---- intra-wave merge (wave32 shuffles) --------------------------------
    #pragma unroll
    for (int o = 16; o >= 1; o >>= 1) {
        const float ms2 = __shfl_xor(m_s, o, 32);
        const float zs2 = __shfl_xor(Z_s, o, 32);
        const float mt2 = __shfl_xor(m_t, o, 32);
        const float zt2 = __shfl_xor(Z_t, o, 32);
        const float A2  = __shfl_xor(A,  o, 32);
        const float B2  = __shfl_xor(Bm, o, 32);
        {
            const float nm = fmaxf(m_s, ms2);
            Z_s = Z_s * __expf(m_s - nm) + zs2 * __expf(ms2 - nm);
            m_s = nm;
        }
        {
            const float nm = fmaxf(m_t, mt2);
            const float r1 = __expf(m_t - nm), r2 = __expf(mt2 - nm);
            Z_t = Z_t * r1 + zt2 * r2;
            A   = A   * r1 + A2  * r2;
            Bm  = Bm  * r1 + B2  * r2;
            m_t = nm;
        }
    }

    const int wid  = (int)(threadIdx.x >> 5);
    const int lane = (int)(threadIdx.x & 31);
    if (lane == 0) {
        red[wid][0] = m_s; red[wid][1] = Z_s;
        red[wid][2] = m_t; red[wid][3] = Z_t;
        red[wid][4] = A;   red[wid][5] = Bm;
    }
    __syncthreads();

    if (threadIdx.x == 0) {
        float gms = red[0][0], gzs = red[0][1];
        float gmt = red[0][2], gzt = red[0][3];
        float gA  = red[0][4], gB  = red[0][5];
        for (int k = 1; k < TPB / 32; ++k) {
            {
                const float nm = fmaxf(gms, red[k][0]);
                gzs = gzs * __expf(gms - nm) + red[k][1] * __expf(red[k][0] - nm);
                gms = nm;
            }
            {
                const float nm = fmaxf(gmt, red[k][2]);
                const float r1 = __expf(gmt - nm), r2 = __expf(red[k][2] - nm);
                gzt = gzt * r1 + red[k][3] * r2;
                gA  = gA  * r1 + red[k][4] * r2;
                gB  = gB  * r1 + red[k][5] * r2;
                gmt = nm;
            }
        }
        // KL = (A + B*((m_s+lnZ_s)-(m_t+lnZ_t))) / Z_t
        const float C    = (gms + __logf(gzs)) - (gmt + __logf(gzt));
        const float loss = (gA + gB * C) / gzt;
        const int label  = gt[b * SEQ + s + 1];
        const float mask = (label != IGNORE_INDEX) ? 1.0f : 0.0f;
        ws[r]         = loss * mask;
        ws[NROWS + r] = mask;
    }
}

__global__ __launch_bounds__(256)
void distill_finalize_kernel(const float* __restrict__ ws, float* __restrict__ out) {
    __shared__ float rl[8], rm[8];
    float ls = 0.0f, ms = 0.0f;
    for (int i = (int)threadIdx.x; i < NROWS; i += 256) {
        ls += ws[i];
        ms += ws[NROWS + i];
    }
    #pragma unroll
    for (int o = 16; o >= 1; o >>= 1) {
        ls += __shfl_xor(ls, o, 32);
        ms += __shfl_xor(ms, o, 32);
    }
    const int wid = (int)(threadIdx.x >> 5), lane = (int)(threadIdx.x & 31);
    if (lane == 0) { rl[wid] = ls; rm[wid] = ms; }
    __syncthreads();
    if (threadIdx.x == 0) {
        float a = 0.0f, m = 0.0f;
        for (int k = 0; k < 8; ++k) { a += rl[k]; m += rm[k]; }
        out[0] = a / m;
    }
}

extern "C" void kernel_launch(void* const* d_in, const int* in_sizes, int n_in,
                              void* d_out, int out_size, void* d_ws, size_t ws_size,
                              hipStream_t stream) {
    (void)in_sizes; (void)n_in; (void)out_size; (void)ws_size;
    const int*   gt   = (const int*)d_in[0];
    const float* slog = (const float*)d_in[1];
    const float* tlog = (const float*)d_in[2];
    float* ws  = (float*)d_ws;
    float* out = (float*)d_out;

    distill_row_kernel<<<NROWS, TPB, 0, stream>>>(gt, slog, tlog, ws);
    distill_finalize_kernel<<<1, 256, 0, stream>>>(ws, out);
}